// PillarLayer_64922725646715
// MI455X (gfx1250) — compile-verified
//
#include <hip/hip_runtime.h>
#include <stdint.h>

// ---------------------------------------------------------------------------
// PointPillars voxelization for MI455X (gfx1250, wave32).
// Memory-bound scatter: no matrix math -> no WMMA. CDNA5-specific paths used:
//   * global_load_async_to_lds_b128 + s_wait_asynccnt (ASYNCcnt) point staging
//   * wave32 __shfl_xor bitonic sort (stable intra-voxel point order)
// Workspace requirement: ~38 MB.
// ---------------------------------------------------------------------------

namespace {
constexpr int       B     = 8;
constexpr int       N     = 250000;
constexpr int       NXv   = 432, NYv = 496, NZv = 1;
constexpr int       NVOX  = NXv * NYv * NZv;              // 214272
constexpr int       MAXV  = 16000;
constexpr int       MAXP  = 32;
constexpr int       SCAN_CHUNK = 1024;
constexpr int       NBLK  = (NVOX + SCAN_CHUNK - 1) / SCAN_CHUNK;  // 210
constexpr long long TOTAL_PTS = (long long)B * N;
} // namespace

__device__ __forceinline__ bool point_to_vid(float x, float y, float z, int& vid) {
  // Match the reference exactly: floor((p - lo) / voxel_size), float32 divide.
  int gx = (int)floorf((x - 0.0f)   / 0.16f);
  int gy = (int)floorf((y + 39.68f) / 0.16f);
  int gz = (int)floorf((z + 3.0f)   / 4.0f);
  vid = (gz * NYv + gy) * NXv + gx;
  return (gx >= 0) & (gx < NXv) & (gy >= 0) & (gy < NYv) & (gz >= 0) & (gz < NZv);
}

// K0: zero counters/cursors/block-sums; init rank->vid map to -1, rank->cnt to 0.
__global__ void __launch_bounds__(256) k_init(int* __restrict__ zbase, long long nz,
                                              int* __restrict__ vor,
                                              int* __restrict__ cor, int nv) {
  long long i = (long long)blockIdx.x * 256 + threadIdx.x;
  if (i < nz) zbase[i] = 0;
  if (i < nv) { vor[i] = -1; cor[i] = 0; }
}

// K1: per-voxel point counts. Points staged global->LDS via CDNA5 async copy.
__global__ void __launch_bounds__(256) k_count(const float* __restrict__ pts,
                                               int* __restrict__ cnt) {
  __shared__ float4 stage[256];
  const int t = threadIdx.x;
  long long gidx = (long long)blockIdx.x * 256 + t;
  long long src  = (gidx < TOTAL_PTS) ? gidx : 0;       // clamp tail, stay uniform
  const float* gptr = pts + src * 4;
  unsigned lds_addr = (unsigned)(uintptr_t)(&stage[t]); // low 32 bits = LDS offset
  asm volatile("global_load_async_to_lds_b128 %0, %1, off"
               :: "v"(lds_addr), "v"(gptr) : "memory");
  asm volatile("s_wait_asynccnt 0" ::: "memory");
  float4 p = stage[t];
  if (gidx >= TOTAL_PTS) return;
  int vid;
  if (!point_to_vid(p.x, p.y, p.z, vid)) return;
  int b = (int)(gidx / N);
  atomicAdd(&cnt[(size_t)b * NVOX + vid], 1);
}

// K2a: per-block occupancy sums (1024 voxels / block).
__global__ void __launch_bounds__(256) k_scan1(const int* __restrict__ cnt,
                                               int* __restrict__ bsums) {
  const int b = blockIdx.y, blk = blockIdx.x, t = threadIdx.x;
  const int base = blk * SCAN_CHUNK + t * 4;
  const int* c = cnt + (size_t)b * NVOX;
  int s = 0;
#pragma unroll
  for (int i = 0; i < 4; ++i) {
    int v = base + i;
    if (v < NVOX) s += (c[v] > 0);
  }
  __shared__ int red[256];
  red[t] = s;
  __syncthreads();
  for (int off = 128; off > 0; off >>= 1) {
    if (t < off) red[t] += red[t + off];
    __syncthreads();
  }
  if (t == 0) bsums[b * NBLK + blk] = red[0];
}

// K2b: tiny serial exclusive scan of the 210 block sums per batch.
__global__ void k_scan2(int* __restrict__ bsums) {
  if (threadIdx.x != 0) return;
  const int b = blockIdx.x;
  int acc = 0;
  for (int i = 0; i < NBLK; ++i) {
    int v = bsums[b * NBLK + i];
    bsums[b * NBLK + i] = acc;
    acc += v;
  }
}

// K2c: final ranks; record rank->vid and rank->count for kept voxels.
__global__ void __launch_bounds__(256) k_scan3(const int* __restrict__ cnt,
                                               const int* __restrict__ bsums,
                                               int* __restrict__ rank,
                                               int* __restrict__ vor,
                                               int* __restrict__ cor) {
  const int b = blockIdx.y, blk = blockIdx.x, t = threadIdx.x;
  const int base = blk * SCAN_CHUNK + t * 4;
  const int* c = cnt + (size_t)b * NVOX;
  int occ[4];
  int s = 0;
#pragma unroll
  for (int i = 0; i < 4; ++i) {
    int v = base + i;
    occ[i] = (v < NVOX) ? (c[v] > 0) : 0;
    s += occ[i];
  }
  __shared__ int sc[256];
  sc[t] = s;
  __syncthreads();
  for (int off = 1; off < 256; off <<= 1) {
    int add = (t >= off) ? sc[t - off] : 0;
    __syncthreads();
    sc[t] += add;
    __syncthreads();
  }
  int pre = bsums[b * NBLK + blk] + sc[t] - s;  // exclusive prefix, this thread's 1st voxel
#pragma unroll
  for (int i = 0; i < 4; ++i) {
    int v = base + i;
    if (v < NVOX && occ[i]) {
      int r = pre++;
      rank[(size_t)b * NVOX + v] = r;
      if (r < MAXV) {
        vor[b * MAXV + r] = v;
        cor[b * MAXV + r] = c[v];
      }
    }
  }
}

// K3: scatter point indices into per-pillar slots (arbitrary slot order,
// fixed up by the sort in K4).
__global__ void __launch_bounds__(256) k_scatter(const float* __restrict__ pts,
                                                 const int* __restrict__ rank,
                                                 int* __restrict__ cursor,
                                                 int* __restrict__ ptidx) {
  long long gidx = (long long)blockIdx.x * 256 + threadIdx.x;
  if (gidx >= TOTAL_PTS) return;
  const float4 p = reinterpret_cast<const float4*>(pts)[gidx];
  int vid;
  if (!point_to_vid(p.x, p.y, p.z, vid)) return;
  int b  = (int)(gidx / N);
  int pi = (int)(gidx - (long long)b * N);
  int r  = rank[(size_t)b * NVOX + vid];
  if ((unsigned)r >= (unsigned)MAXV) return;
  int slot = atomicAdd(&cursor[(size_t)b * NVOX + vid], 1);
  if (slot < MAXP) ptidx[((size_t)b * MAXV + r) * MAXP + slot] = pi;
}

// K4: one wave32 per output pillar. Bitonic-sort point indices (restores the
// reference's stable original-index order => deterministic output), gather
// points (L2-resident: 32 MB points << 192 MB L2), write all 32 rows + meta.
__global__ void __launch_bounds__(256) k_write(const float* __restrict__ pts,
                                               const int* __restrict__ vor,
                                               const int* __restrict__ cor,
                                               const int* __restrict__ ptidx,
                                               float* __restrict__ pillars,
                                               int* __restrict__ coors,
                                               int* __restrict__ npts) {
  const int lane = threadIdx.x & 31;
  const int wid  = threadIdx.x >> 5;
  const int v    = blockIdx.x * 8 + wid;     // [0, B*MAXV), exact coverage
  const int b    = v / MAXV;
  const int vid  = vor[v];
  const int c    = cor[v];
  const int m    = (vid >= 0) ? min(c, MAXP) : 0;

  float4 out = make_float4(0.f, 0.f, 0.f, 0.f);
  if (m > 0) {
    int key = (lane < m) ? ptidx[(size_t)v * MAXP + lane] : 0x7FFFFFFF;
#pragma unroll
    for (int k = 2; k <= 32; k <<= 1) {
#pragma unroll
      for (int j = k >> 1; j > 0; j >>= 1) {
        int partner = __shfl_xor(key, j, 32);
        const bool lower = ((lane & j) == 0);
        const bool asc   = ((lane & k) == 0);
        key = (lower == asc) ? min(key, partner) : max(key, partner);
      }
    }
    if (lane < m)
      out = reinterpret_cast<const float4*>(pts)[(long long)b * N + key];
  }
  reinterpret_cast<float4*>(pillars)[(size_t)v * MAXP + lane] = out;

  if (lane == 0) {
    int4 co;
    co.x = b;
    if (m > 0) {
      int gz  = vid / (NXv * NYv);
      int rem = vid % (NXv * NYv);
      co.y = rem % NXv;
      co.z = rem / NXv;
      co.w = gz;
    } else {
      co.y = co.z = co.w = -1;
    }
    reinterpret_cast<int4*>(coors)[v] = co;
    npts[v] = m;
  }
}

extern "C" void kernel_launch(void* const* d_in, const int* in_sizes, int n_in,
                              void* d_out, int out_size, void* d_ws, size_t ws_size,
                              hipStream_t stream) {
  (void)in_sizes; (void)n_in; (void)out_size; (void)ws_size;
  const float* pts = (const float*)d_in[0];        // [B, N, 4] f32

  // Output: pillars f32 [B*MAXV, 32, 4] | coors i32 [B*MAXV, 4] | npts i32 [B*MAXV]
  float* pillars = (float*)d_out;
  int*   coors   = (int*)((float*)d_out + (size_t)B * MAXV * MAXP * 4);
  int*   npts    = (int*)((float*)d_out + (size_t)B * MAXV * MAXP * 4 + (size_t)B * MAXV * 4);

  // Workspace layout (ints). Total ~38 MB.
  int* cnt    = (int*)d_ws;
  int* cursor = cnt    + (size_t)B * NVOX;
  int* bsums  = cursor + (size_t)B * NVOX;
  int* rank   = bsums  + (size_t)B * NBLK;
  int* vor    = rank   + (size_t)B * NVOX;
  int* cor    = vor    + (size_t)B * MAXV;
  int* ptidx  = cor    + (size_t)B * MAXV;

  const long long nzero = (long long)2 * B * NVOX + (long long)B * NBLK;  // cnt+cursor+bsums
  {
    int blocks = (int)((nzero + 255) / 256);
    k_init<<<blocks, 256, 0, stream>>>(cnt, nzero, vor, cor, B * MAXV);
  }
  {
    int blocks = (int)((TOTAL_PTS + 255) / 256);
    k_count<<<blocks, 256, 0, stream>>>(pts, cnt);
  }
  k_scan1<<<dim3(NBLK, B), 256, 0, stream>>>(cnt, bsums);
  k_scan2<<<B, 32, 0, stream>>>(bsums);
  k_scan3<<<dim3(NBLK, B), 256, 0, stream>>>(cnt, bsums, rank, vor, cor);
  {
    int blocks = (int)((TOTAL_PTS + 255) / 256);
    k_scatter<<<blocks, 256, 0, stream>>>(pts, rank, cursor, ptidx);
  }
  k_write<<<(B * MAXV) / 8, 256, 0, stream>>>(pts, vor, cor, ptidx, pillars, coors, npts);
}